// CustomStyleLoss_23021024706837
// MI455X (gfx1250) — compile-verified
//
#include <hip/hip_runtime.h>
#include <hip/hip_bf16.h>

// CustomStyleLoss on MI455X (gfx1250, wave32).
// loss = sum_r mean_c ( (mean_{chunk c} in - mean_{chunk c} style)^2 )
//      = (1/25) * sum_{r,c} d[r,c]^2,   d[r,c] = 0.01 * sum_{j in chunk}(in - style)
//
// Memory-bound: 163.8 MB streamed once -> ~7.0 us floor @ 23.3 TB/s.
// Segment sums ride the matrix pipe via V_WMMA_F32_16X16X4_F32:
//   A = 16 rows x 4 consecutive diff elements (LDS, double-buffered)
//   B[k,n] = 0.01 * (n == chunk_column)   (constant along K -> layout-proof)
// Tile is zero-padded from 100 to 128 floats/row so the 32 k-steps split
// exactly 4-per-wave: constant trip counts, no exec-masked WMMA.

typedef __attribute__((ext_vector_type(2))) float v2f;
typedef __attribute__((ext_vector_type(8))) float v8f;

#define ROWS_PER_BLK 16
#define ROW_LEN      2500   // 25 chunks * 100
#define N_CHUNKS     25
#define Q_PER_CHUNK  25     // real float4 per chunk
#define Q_PAD        32     // staged float4 per row (128 floats, zero-padded)
#define PAD          132    // LDS row stride in floats: banks {4L..4L+3} distinct
#define KSTEPS_PER_WAVE 4   // 32 k-steps / 8 waves

__device__ __forceinline__ void stage_tile(float* __restrict__ dst,
                                           const float4* __restrict__ in4,
                                           const float4* __restrict__ st4,
                                           int rowbase, int c, int tid, bool pf) {
    // 16 rows x 32 float4 slots = 512 = exactly 2 per thread (uniform flow).
    #pragma unroll
    for (int u = 0; u < 2; ++u) {
        const int idx = tid + u * 256;
        const int r = idx >> 5;        // 0..15
        const int q = idx & 31;        // 0..31
        float* d = &dst[r * PAD + q * 4];
        if (q < Q_PER_CHUNK) {
            const int g = (rowbase + r) * (ROW_LEN / 4) + c * Q_PER_CHUNK + q;
            float4 a = in4[g];
            float4 s = st4[g];
            if (pf) {                  // warm chunk c+1 (double buffer covers it,
                __builtin_prefetch(&in4[g + Q_PER_CHUNK], 0, 0);  // prefetch pulls
                __builtin_prefetch(&st4[g + Q_PER_CHUNK], 0, 0);  // it earlier)
            }
            d[0] = a.x - s.x; d[1] = a.y - s.y;
            d[2] = a.z - s.z; d[3] = a.w - s.w;
        } else {
            d[0] = 0.0f; d[1] = 0.0f; d[2] = 0.0f; d[3] = 0.0f;
        }
    }
}

__device__ __forceinline__ v8f chunk_wmma(const float* __restrict__ ap,
                                          v2f b, v8f acc, int wave) {
    // Exactly 4 k-steps per wave, fully unrolled: ds_load_b64 -> v_wmma,
    // constant trip count => no exec masking around WMMA.
    #pragma unroll
    for (int j = 0; j < KSTEPS_PER_WAVE; ++j) {
        const float* p = ap + (wave * KSTEPS_PER_WAVE + j) * 4;
        v2f a; a[0] = p[0]; a[1] = p[1];
        acc = __builtin_amdgcn_wmma_f32_16x16x4_f32(
            false, a, false, b, (short)0, acc, false, false);
    }
    return acc;
}

__global__ __launch_bounds__(256)
void style_loss_wmma_kernel(const float* __restrict__ input,
                            const float* __restrict__ style,
                            float* __restrict__ block_part) {
    __shared__ float sdiff[2][ROWS_PER_BLK * PAD];  // double-buffered diff tile
    __shared__ float sred[8 * 512];                 // per-wave accumulator dump
    __shared__ float rbuf[256];                     // block tree-reduce

    const int tid  = threadIdx.x;
    const int lane = tid & 31;
    const int wave = tid >> 5;
    const int rowbase = blockIdx.x * ROWS_PER_BLK;

    const float4* in4 = (const float4*)input;
    const float4* st4 = (const float4*)style;

    // A-operand addressing for V_WMMA_F32_16X16X4_F32 (wave32 layout):
    // lanes 0-15 supply K=0,1 of row=lane; lanes 16-31 supply K=2,3 of row=lane-16.
    const int rowsel = (lane < 16) ? lane : (lane - 16);
    const int koff   = (lane < 16) ? 0 : 2;
    const int abase  = rowsel * PAD + koff;

    v8f acc0 = {};   // chunk columns 0..15
    v8f acc1 = {};   // chunk columns 16..24 (cols 9..15 stay zero)

    stage_tile(sdiff[0], in4, st4, rowbase, 0, tid, true);
    __syncthreads();

    // Chunks 0..15 -> acc0 (no accumulator branch anywhere).
    for (int c = 0; c < 16; ++c) {
        const float* cur = sdiff[c & 1];
        float*       nxt = sdiff[(c & 1) ^ 1];
        stage_tile(nxt, in4, st4, rowbase, c + 1, tid, c + 2 < N_CHUNKS);
        const float bv = ((lane & 15) == c) ? 0.01f : 0.0f;
        v2f b; b[0] = bv; b[1] = bv;
        acc0 = chunk_wmma(&cur[abase], b, acc0, wave);
        __syncthreads();
    }
    // Chunks 16..24 -> acc1.
    for (int c = 16; c < N_CHUNKS; ++c) {
        const float* cur = sdiff[c & 1];
        float*       nxt = sdiff[(c & 1) ^ 1];
        if (c + 1 < N_CHUNKS)
            stage_tile(nxt, in4, st4, rowbase, c + 1, tid, c + 2 < N_CHUNKS);
        const float bv = ((lane & 15) == (c - 16)) ? 0.01f : 0.0f;
        v2f b; b[0] = bv; b[1] = bv;
        acc1 = chunk_wmma(&cur[abase], b, acc1, wave);
        __syncthreads();
    }

    // Combine the 8 per-wave partial accumulators BEFORE squaring.
    {
        const int base = wave * 512 + lane * 16;
        #pragma unroll
        for (int j = 0; j < 8; ++j) {
            sred[base + j]     = acc0[j];
            sred[base + 8 + j] = acc1[j];
        }
    }
    __syncthreads();

    float part = 0.0f;
    #pragma unroll
    for (int u = 0; u < 2; ++u) {
        const int s = tid + u * 256;
        float d = 0.0f;
        #pragma unroll
        for (int w = 0; w < 8; ++w) d += sred[w * 512 + s];
        part += d * d;                       // zero slots contribute zero
    }

    rbuf[tid] = part;
    __syncthreads();
    #pragma unroll
    for (int s = 128; s > 0; s >>= 1) {
        if (tid < s) rbuf[tid] += rbuf[tid + s];
        __syncthreads();
    }
    if (tid == 0) block_part[blockIdx.x] = rbuf[0];
}

__global__ __launch_bounds__(512)
void style_loss_final_reduce(const float* __restrict__ part,
                             float* __restrict__ out, int n) {
    __shared__ float buf[512];
    float v = 0.0f;
    for (int i = threadIdx.x; i < n; i += 512) v += part[i];
    buf[threadIdx.x] = v;
    __syncthreads();
    #pragma unroll
    for (int s = 256; s > 0; s >>= 1) {
        if (threadIdx.x < s) buf[threadIdx.x] += buf[threadIdx.x + s];
        __syncthreads();
    }
    if (threadIdx.x == 0) out[0] = buf[0] * (1.0f / 25.0f);
}

extern "C" void kernel_launch(void* const* d_in, const int* in_sizes, int n_in,
                              void* d_out, int out_size, void* d_ws, size_t ws_size,
                              hipStream_t stream) {
    const float* input = (const float*)d_in[0];
    const float* style = (const float*)d_in[1];
    float* out = (float*)d_out;
    float* ws  = (float*)d_ws;

    const int total  = in_sizes[0];           // 16*512*50*50 = 20,480,000
    const int n_rows = total / ROW_LEN;       // 8192
    const int nblk   = n_rows / ROWS_PER_BLK; // 512

    style_loss_wmma_kernel<<<nblk, 256, 0, stream>>>(input, style, ws);
    style_loss_final_reduce<<<1, 512, 0, stream>>>(ws, out, nblk);
}